// Isomorphic_Feature_Extraction_38981123178956
// MI455X (gfx1250) — compile-verified
//
#include <hip/hip_runtime.h>

typedef float v2f __attribute__((ext_vector_type(2)));
typedef float v8f __attribute__((ext_vector_type(8)));

#define B_    4
#define H_    64
#define NH    60          // window positions per image dim
#define NWIN  3600        // 60*60
#define M_TOT 14400       // B_*NWIN
#define C_    32
#define P_    120
#define PPAD  128
#define NCOL  4096        // C_*PPAD
#define K_    25
#define KP    28          // K padded to multiple of 4
#define K2    14          // KP/2 (float2 pairs)

// ---------------------------------------------------------------------------
// Pack sliding windows -> Apack[m][0..27] (row-major, zero padded), nx[m]
// ---------------------------------------------------------------------------
__global__ void pack_sub_kernel(const float* __restrict__ x,
                                float* __restrict__ Apack,
                                float* __restrict__ nx) {
    int m = blockIdx.x * blockDim.x + threadIdx.x;
    if (m >= M_TOT) return;
    int b  = m / NWIN;
    int n  = m - b * NWIN;
    int nh = n / NH;
    int nw = n - nh * NH;
    const float* xb = x + b * H_ * H_;
    float w[KP];
    float s2 = 0.f;
#pragma unroll
    for (int i = 0; i < 5; ++i)
#pragma unroll
        for (int j = 0; j < 5; ++j) {
            float v = xb[(nh + i) * H_ + (nw + j)];
            w[i * 5 + j] = v;
            s2 += v * v;
        }
    w[25] = 0.f; w[26] = 0.f; w[27] = 0.f;
    float* ap = Apack + m * KP;
#pragma unroll
    for (int k = 0; k < KP; ++k) ap[k] = w[k];
    nx[m] = s2;
}

// ---------------------------------------------------------------------------
// Build B operand: Bpack[k2][col] = (Amat[2k2,col], Amat[2k2+1,col]) as float2
// col = c*128 + p ; Amat[c,p,i,j] = kernel1[c, perm_p(i), perm_p(j)]
// Padded p (>=120): zero matrix, nA = 1e30 sentinel (never wins the min)
// ---------------------------------------------------------------------------
__global__ void pack_A_kernel(const float* __restrict__ kernel1,
                              const float* __restrict__ P,
                              float* __restrict__ Bpack,
                              float* __restrict__ nA) {
    int col = blockIdx.x * blockDim.x + threadIdx.x;
    if (col >= NCOL) return;
    int c = col >> 7;
    int p = col & (PPAD - 1);
    float amat[KP];
    float s = 0.f;
    if (p < P_) {
        int perm[5];
        const float* Pp = P + p * 25;
#pragma unroll
        for (int i = 0; i < 5; ++i) {
            int best = 0; float bv = Pp[i * 5 + 0];
#pragma unroll
            for (int k = 1; k < 5; ++k) {
                float v = Pp[i * 5 + k];
                if (v > bv) { bv = v; best = k; }
            }
            perm[i] = best;
        }
        const float* Kc = kernel1 + c * 25;
#pragma unroll
        for (int i = 0; i < 5; ++i)
#pragma unroll
            for (int j = 0; j < 5; ++j) {
                float v = Kc[perm[i] * 5 + perm[j]];
                amat[i * 5 + j] = v;
                s += v * v;
            }
        amat[25] = 0.f; amat[26] = 0.f; amat[27] = 0.f;
    } else {
#pragma unroll
        for (int k = 0; k < KP; ++k) amat[k] = 0.f;
        s = 1.0e30f;
    }
#pragma unroll
    for (int k2 = 0; k2 < K2; ++k2) {
        Bpack[(k2 * NCOL + col) * 2 + 0] = amat[2 * k2 + 0];
        Bpack[(k2 * NCOL + col) * 2 + 1] = amat[2 * k2 + 1];
    }
    nA[col] = s;
}

// ---------------------------------------------------------------------------
// GEMM (f32 WMMA 16x16x4) + fused  min_p( nA - 2*cross ) ;  feat = nx + min
// One wave per 16-row M tile; 4 waves per block.
// ---------------------------------------------------------------------------
__global__ void gemm_min_kernel(const float* __restrict__ Apack,
                                const float* __restrict__ Bpack,
                                const float* __restrict__ nx,
                                const float* __restrict__ nA,
                                float* __restrict__ feat) {
    const int lane  = threadIdx.x & 31;
    const int wave  = threadIdx.x >> 5;
    const int mtile = blockIdx.x * 4 + wave;
    const int mbase = mtile * 16;
    const int hi    = lane >> 4;      // half-wave select (K sub-pair / M+8)
    const int l15   = lane & 15;

    const v2f* Ap = (const v2f*)Apack;
    const v2f* Bp = (const v2f*)Bpack;

    // A operand: 16 rows x 28 K, resident in registers, reused for all tiles.
    v2f areg[7];
#pragma unroll
    for (int s = 0; s < 7; ++s)
        areg[s] = Ap[(mbase + l15) * K2 + 2 * s + hi];

    for (int c = 0; c < C_; ++c) {
        v8f minv;
#pragma unroll
        for (int r = 0; r < 8; ++r) minv[r] = 3.0e38f;

        for (int t = 0; t < 8; ++t) {           // 8 p-tiles of 16 (120 -> 128)
            const int colbase = c * PPAD + t * 16;
            const int col     = colbase + l15;
            v8f acc = {};
#pragma unroll
            for (int s = 0; s < 7; ++s) {       // K = 28 in steps of 4
                v2f b = Bp[(2 * s + hi) * NCOL + col];
                acc = __builtin_amdgcn_wmma_f32_16x16x4_f32(
                        false, areg[s], false, b, (short)0, acc, false, false);
            }
            float na = nA[col];
#pragma unroll
            for (int r = 0; r < 8; ++r)
                minv[r] = fminf(minv[r], na - 2.0f * acc[r]);
        }

        // min across the 16-lane N groups; lane l15==0 of each half holds it.
#pragma unroll
        for (int r = 0; r < 8; ++r) {
            float v = minv[r];
            v = fminf(v, __shfl_xor(v, 1, 32));
            v = fminf(v, __shfl_xor(v, 2, 32));
            v = fminf(v, __shfl_xor(v, 4, 32));
            v = fminf(v, __shfl_xor(v, 8, 32));
            if (l15 == 0) {
                int m = mbase + r + 8 * hi;     // C/D layout: M = r + 8*hi
                feat[m * C_ + c] = nx[m] + v;
            }
        }
    }
}

// ---------------------------------------------------------------------------
// Softmax over n (3600) for each (b,c); out[b, n*32 + c]
// ---------------------------------------------------------------------------
__global__ void softmax_n_kernel(const float* __restrict__ feat,
                                 float* __restrict__ out) {
    const int b = blockIdx.x >> 5;
    const int c = blockIdx.x & 31;
    const int tid = threadIdx.x;
    __shared__ float red[256];

    const float* fb = feat + (size_t)b * NWIN * C_ + c;

    float mx = -3.0e38f;
    for (int n = tid; n < NWIN; n += 256)
        mx = fmaxf(mx, -fb[n * C_]);
    red[tid] = mx;
    __syncthreads();
    for (int s = 128; s > 0; s >>= 1) {
        if (tid < s) red[tid] = fmaxf(red[tid], red[tid + s]);
        __syncthreads();
    }
    mx = red[0];
    __syncthreads();

    float sum = 0.f;
    for (int n = tid; n < NWIN; n += 256)
        sum += __expf(-fb[n * C_] - mx);
    red[tid] = sum;
    __syncthreads();
    for (int s = 128; s > 0; s >>= 1) {
        if (tid < s) red[tid] += red[tid + s];
        __syncthreads();
    }
    const float inv = 1.0f / red[0];

    float* ob = out + (size_t)b * NWIN * C_ + c;
    for (int n = tid; n < NWIN; n += 256)
        ob[n * C_] = __expf(-fb[n * C_] - mx) * inv;
}

// ---------------------------------------------------------------------------
extern "C" void kernel_launch(void* const* d_in, const int* in_sizes, int n_in,
                              void* d_out, int out_size, void* d_ws, size_t ws_size,
                              hipStream_t stream) {
    const float* x       = (const float*)d_in[0];   // (4,64,64)
    const float* kernel1 = (const float*)d_in[1];   // (32,5,5)
    const float* P       = (const float*)d_in[2];   // (120,5,5)

    float* ws    = (float*)d_ws;
    float* Apack = ws;                        // 14400*28      = 403200
    float* Bpack = Apack + (size_t)M_TOT*KP;  // 14*4096*2     = 114688
    float* nx    = Bpack + (size_t)K2*NCOL*2; // 14400
    float* nA    = nx + M_TOT;                // 4096
    float* feat  = nA + NCOL;                 // 14400*32      = 460800

    pack_sub_kernel<<<(M_TOT + 255) / 256, 256, 0, stream>>>(x, Apack, nx);
    pack_A_kernel<<<NCOL / 256, 256, 0, stream>>>(kernel1, P, Bpack, nA);
    gemm_min_kernel<<<M_TOT / 64, 128, 0, stream>>>(Apack, Bpack, nx, nA, feat);
    softmax_n_kernel<<<B_ * C_, 256, 0, stream>>>(feat, (float*)d_out);
}